// DiffusionModel1d_40338332844168
// MI455X (gfx1250) — compile-verified
//
#include <hip/hip_runtime.h>

#define BATCH 8192
#define MROW  2048
#define NSZ   2047
#define CHUNK 32
#define NT    64            // ceil(2047/32)
#define PADW  33            // tile row stride in floats (odd -> conflict-free)

typedef unsigned int u32x4 __attribute__((ext_vector_type(4)));
typedef int          i32x8 __attribute__((ext_vector_type(8)));
typedef int          i32x4 __attribute__((ext_vector_type(4)));

__device__ __forceinline__ float fast_rcp(float d) {
  float r = __builtin_amdgcn_rcpf(d);
  float e = __builtin_fmaf(-d, r, 1.0f);
  return __builtin_fmaf(r, e, r);   // 1 Newton step -> ~0.5 ulp
}

// One TDM load: 32 rows x 33 cols of f32 from x[b0..b0+31][j0..j0+32]
// (row pitch 2048) packed contiguously into LDS at lds_off (row stride 33 floats).
// D# per CDNA5 ISA 8.3/8.4.  tensor_dim0 = 2048 - j0 => zero-fill past row edge.
__device__ __forceinline__ void tdm_load_tile(const float* x, unsigned b0,
                                              int t, unsigned lds_off) {
  asm volatile("" ::: "memory");    // keep issue ordered vs prior LDS use
  const unsigned j0 = (unsigned)t * CHUNK;
  unsigned long long ga =
      (unsigned long long)(const void*)x +
      (((unsigned long long)b0 * (unsigned long long)MROW + j0) << 2);
  const unsigned dim0 = (unsigned)MROW - j0;

  u32x4 g0;
  g0.x = 1u;                                           // count=1 valid user desc
  g0.y = lds_off;                                      // lds_addr (bytes)
  g0.z = (unsigned)(ga & 0xffffffffull);               // global_addr[31:0]
  g0.w = (unsigned)((ga >> 32) & 0x01ffffffull)        // global_addr[56:32]
         | 0x80000000u;                                // type=2 ("image")

  i32x8 g1;
  g1[0] = (int)(2u << 16);                             // data_size=2 (4B)
  g1[1] = (int)(dim0 << 16);                           // tensor_dim0[15:0] @ bit48
  g1[2] = (int)((dim0 >> 16) | ((unsigned)CHUNK << 16)); // dim0 hi | tensor_dim1=32
  g1[3] = (int)((unsigned)PADW << 16);                 // tile_dim0=33 @ bit112
  g1[4] = (int)CHUNK;                                  // tile_dim1=32 ; tile_dim2=0
  g1[5] = (int)MROW;                                   // tensor_dim0_stride=2048
  g1[6] = 0;
  g1[7] = 0;
  i32x4 z4; z4[0] = 0; z4[1] = 0; z4[2] = 0; z4[3] = 0;
  i32x8 z8; z8[0] = 0; z8[1] = 0; z8[2] = 0; z8[3] = 0;
            z8[4] = 0; z8[5] = 0; z8[6] = 0; z8[7] = 0;
  __builtin_amdgcn_tensor_load_to_lds(g0, g1, z4, z4, z8, 0);
}

// ---- repeat helpers: force straight-line code + named scalar registers -----
#define REP31(X) X(0) X(1) X(2) X(3) X(4) X(5) X(6) X(7) X(8) X(9) X(10) X(11) \
  X(12) X(13) X(14) X(15) X(16) X(17) X(18) X(19) X(20) X(21) X(22) X(23) \
  X(24) X(25) X(26) X(27) X(28) X(29) X(30)
#define REP32(X) REP31(X) X(31)
#define REP31_REV(X) X(30) X(29) X(28) X(27) X(26) X(25) X(24) X(23) X(22) \
  X(21) X(20) X(19) X(18) X(17) X(16) X(15) X(14) X(13) X(12) X(11) X(10) \
  X(9) X(8) X(7) X(6) X(5) X(4) X(3) X(2) X(1) X(0)
#define REP32_REV(X) X(31) REP31_REV(X)

#define DECLV(J) float cR##J, uR##J;
// generic forward step (interior element): keeps c/u in named registers
#define FSTEP(J) { const float Kj1 = __expf(tp[tbase + (J) + 1]);              \
    const float denom = __builtin_fmaf(Kj, cp, Kj + Kj1);                      \
    const float r = fast_rcp(denom);                                           \
    cp = -Kj1 * r;                                                             \
    up = __builtin_fmaf(Kj, up, f_lds[jbase + (J)]) * r;                       \
    cR##J = cp; uR##J = up; Kj = Kj1; }
// last element (j == 2046): K[2047] := 0
#define FSTEP_LAST(J) { const float denom = __builtin_fmaf(Kj, cp, Kj);        \
    const float r = fast_rcp(denom);                                           \
    cp = 0.0f;                                                                 \
    up = __builtin_fmaf(Kj, up, f_lds[jbase + (J)]) * r;                       \
    cR##J = 0.0f; uR##J = up; Kj = 0.0f; }
// backward step: u_out = u - c*u_next, stage h2*u_out into the tile buffer
#define BSTEP(J) { const float v = __builtin_fmaf(-cR##J, u_next, uR##J);      \
    u_next = v; tp[tbase + (J)] = H2 * v; }

__global__ __launch_bounds__(32)
void thomas_tdm_kernel(const float* __restrict__ x, const float* __restrict__ f,
                       float* __restrict__ out) {
  __shared__ float f_lds[NSZ + 1];
  __shared__ float tile[2][CHUNK * PADW];

  const int lane = threadIdx.x;               // system within this wave
  const unsigned b0 = blockIdx.x * 32u;       // first system of this wave
  const float H2 = (float)(1.0 / ((double)NSZ * (double)NSZ));

  const unsigned toff0 = (unsigned)(unsigned long long)&tile[0][0];
  const unsigned toff1 = (unsigned)(unsigned long long)&tile[1][0];
  const int tbase = lane * PADW;

  // Prefetch chunk 0 via TDM, then stage f into LDS while it flies.
  tdm_load_tile(x, b0, 0, toff0);
  for (int i = lane; i < NSZ; i += 32) f_lds[i] = f[i];

  // Per-lane chunk-boundary checkpoints (dynamic index -> scratch, 512B/thread).
  float chkC[NT], chkU[NT];

  // ---------------- Phase 1: forward sweep, checkpoints only ----------------
  // Folded init: c_prev=1 makes j=0 identical to the generic step;
  // K[2047]:=0 makes j=2046 (u_last) identical too.
  float cp = 1.0f, up = 0.0f;

  for (int t = 0; t < NT - 1; ++t) {          // chunks 0..62: fully regular
    tdm_load_tile(x, b0, t + 1, ((t + 1) & 1) ? toff1 : toff0);
    __builtin_amdgcn_s_wait_tensorcnt(1);     // oldest (chunk t) complete
    asm volatile("" ::: "memory");
    const float* tp = &tile[t & 1][0];
    chkC[t] = cp; chkU[t] = up;
    const int jbase = t * CHUNK;
    float Kj = __expf(tp[tbase]);
    for (int jj = 0; jj < CHUNK; ++jj) {
      const float Kj1 = __expf(tp[tbase + jj + 1]);
      const float denom = __builtin_fmaf(Kj, cp, Kj + Kj1);
      const float r = fast_rcp(denom);
      cp = -Kj1 * r;
      up = __builtin_fmaf(Kj, up, f_lds[jbase + jj]) * r;
      Kj = Kj1;
    }
  }
  {                                           // peeled chunk 63: j 2016..2046
    __builtin_amdgcn_s_wait_tensorcnt(0);
    asm volatile("" ::: "memory");
    const float* tp = &tile[(NT - 1) & 1][0];
    chkC[NT - 1] = cp; chkU[NT - 1] = up;
    const int jbase = (NT - 1) * CHUNK;
    float Kj = __expf(tp[tbase]);
    for (int jj = 0; jj < CHUNK - 2; ++jj) {  // j 2016..2045
      const float Kj1 = __expf(tp[tbase + jj + 1]);
      const float denom = __builtin_fmaf(Kj, cp, Kj + Kj1);
      const float r = fast_rcp(denom);
      cp = -Kj1 * r;
      up = __builtin_fmaf(Kj, up, f_lds[jbase + jj]) * r;
      Kj = Kj1;
    }
    // j = 2046 (K[2047] := 0) -- value not needed further in phase 1
  }

  // ---------------- Phase 2: recompute per chunk + backward sweep -----------
  float u_next = 0.0f;
  tdm_load_tile(x, b0, NT - 1, ((NT - 1) & 1) ? toff1 : toff0);

  {                                           // peeled chunk 63 first
    tdm_load_tile(x, b0, NT - 2, ((NT - 2) & 1) ? toff1 : toff0);
    __builtin_amdgcn_s_wait_tensorcnt(1);
    asm volatile("" ::: "memory");
    float* tp = &tile[(NT - 1) & 1][0];
    const int jbase = (NT - 1) * CHUNK;
    REP31(DECLV)
    { float cp = chkC[NT - 1], up = chkU[NT - 1], Kj = __expf(tp[tbase]);
      // forward: jj 0..29 generic, jj 30 is j==2046
      FSTEP(0) FSTEP(1) FSTEP(2) FSTEP(3) FSTEP(4) FSTEP(5) FSTEP(6) FSTEP(7)
      FSTEP(8) FSTEP(9) FSTEP(10) FSTEP(11) FSTEP(12) FSTEP(13) FSTEP(14)
      FSTEP(15) FSTEP(16) FSTEP(17) FSTEP(18) FSTEP(19) FSTEP(20) FSTEP(21)
      FSTEP(22) FSTEP(23) FSTEP(24) FSTEP(25) FSTEP(26) FSTEP(27) FSTEP(28)
      FSTEP(29) FSTEP_LAST(30)
      (void)cp; (void)up; (void)Kj;
    }
    REP31_REV(BSTEP)
    asm volatile("" ::: "memory");
    const int jcol = jbase + lane;            // valid only for lane < 31
    for (int r = 0; r < CHUNK; ++r) {
      if (jcol < NSZ)
        out[(unsigned long long)(b0 + (unsigned)r) * NSZ + (unsigned)jcol] =
            tp[r * PADW + lane];
    }
  }

  for (int t = NT - 2; t >= 0; --t) {         // chunks 62..0: fully regular
    if (t > 0) {
      tdm_load_tile(x, b0, t - 1, ((t - 1) & 1) ? toff1 : toff0);
      __builtin_amdgcn_s_wait_tensorcnt(1);
    } else {
      __builtin_amdgcn_s_wait_tensorcnt(0);
    }
    asm volatile("" ::: "memory");
    float* tp = &tile[t & 1][0];
    const int jbase = t * CHUNK;
    REP32(DECLV)
    { float cp = chkC[t], up = chkU[t], Kj = __expf(tp[tbase]);
      REP32(FSTEP)
      (void)cp; (void)up; (void)Kj;
    }
    REP32_REV(BSTEP)
    asm volatile("" ::: "memory");
    const int jcol = jbase + lane;            // always valid in regular chunks
    for (int r = 0; r < CHUNK; ++r) {
      out[(unsigned long long)(b0 + (unsigned)r) * NSZ + (unsigned)jcol] =
          tp[r * PADW + lane];
    }
  }
}

extern "C" void kernel_launch(void* const* d_in, const int* in_sizes, int n_in,
                              void* d_out, int out_size, void* d_ws, size_t ws_size,
                              hipStream_t stream) {
  (void)in_sizes; (void)n_in; (void)out_size; (void)d_ws; (void)ws_size;
  const float* x = (const float*)d_in[0];
  const float* f = (const float*)d_in[1];
  float* out = (float*)d_out;
  dim3 grid(BATCH / 32);
  dim3 block(32);
  hipLaunchKernelGGL(thomas_tdm_kernel, grid, block, 0, stream, x, f, out);
}